// SynoBlock_42734924595781
// MI455X (gfx1250) — compile-verified
//
#include <hip/hip_runtime.h>
#include <hip/hip_bf16.h>
#include <math.h>

// ---------------------------------------------------------------------------
// SynoBlock for MI455X (gfx1250, wave32). fp32 end-to-end using
// V_WMMA_F32_16X16X4_F32 for all matrix products. Problem is HBM-bound
// (~1.25 GB traffic vs ~15 GFLOP), so each input element is read once.
// ---------------------------------------------------------------------------

typedef __attribute__((ext_vector_type(2))) float v2f;
typedef __attribute__((ext_vector_type(8))) float v8f;

#define NB 16
#define NT 32
#define NP 14
#define NH 12
#define ND 64
#define NL 196
#define NDM 768
#define NS 16

__device__ __forceinline__ v8f wmma_f32_k4(v2f a, v2f b, v8f c) {
  // D = A(16x4,f32) * B(4x16,f32) + C(16x16,f32)
  return __builtin_amdgcn_wmma_f32_16x16x4_f32(false, a, false, b, (short)0, c,
                                               false, false);
}

// ===========================================================================
// Kernel 1: temporal Gram + softmax + t_conv.  One block per (b,l).
// Dynamic LDS: affbuf[24][32][32] | Xbuf[2][32][68] | wsm[217]
// ===========================================================================
#define XPAD 68

extern "C" __global__ __launch_bounds__(256) void syno_temporal_kernel(
    const float* __restrict__ q, const float* __restrict__ k,
    const float* __restrict__ tw, const float* __restrict__ tb,
    float* __restrict__ affout) {
  extern __shared__ float smem[];
  float* affbuf = smem;                    // 24*32*32 = 24576
  float* Xbuf   = affbuf + 24 * 32 * 32;   // 2*32*68  = 4352
  float* wsm    = Xbuf + 2 * 32 * XPAD;    // 217

  const int bl  = blockIdx.x;
  const int b   = bl / NL;
  const int l   = bl % NL;
  const int tid = threadIdx.x;
  const int lane = tid & 31;
  const int wave = tid >> 5;
  const int lane15 = lane & 15;
  const int hi = lane >> 4;  // 0 for lanes 0-15, 1 for lanes 16-31

  if (tid < 216) wsm[tid] = tw[tid];
  if (tid == 216) wsm[216] = tb[0];

  // 12 pairs of channels; channels 0..11 <- q, 12..23 <- k, head = c % 12
  for (int cp = 0; cp < 12; ++cp) {
    const int c0 = cp * 2;
    // ---- stage X for both channels of this pair (coalesced over d) ----
    for (int i = tid; i < 2 * NT * ND; i += 256) {
      const int ch = i >> 11;          // which of the 2 channels
      const int r  = (i >> 6) & 31;    // t index
      const int d  = i & 63;           // d index
      const int c  = c0 + ch;
      const float* src = (c < NH) ? q : k;
      const int h = (c < NH) ? c : (c - NH);
      const size_t gi =
          ((size_t)(b * NT + r) * (NL + 1) + (size_t)(l + 1)) * (size_t)NDM +
          (size_t)h * ND + d;
      Xbuf[ch * 32 * XPAD + r * XPAD + d] = src[gi];
    }
    __syncthreads();

    // ---- WMMA: wave -> (channel, 16x16 tile of the 32x32 Gram) ----
    {
      const int ch   = wave >> 2;
      const int tile = wave & 3;
      const int m0 = (tile >> 1) * 16;
      const int n0 = (tile & 1) * 16;
      const float* X = Xbuf + ch * 32 * XPAD;
      const int koffBase = hi ? 2 : 0;
      const float* arow = X + (m0 + lane15) * XPAD + koffBase;
      const float* brow = X + (n0 + lane15) * XPAD + koffBase;
      v8f acc = {};
#pragma unroll
      for (int kb = 0; kb < ND; kb += 4) {
        v2f a;  a.x = arow[kb]; a.y = arow[kb + 1];
        v2f bb; bb.x = brow[kb]; bb.y = brow[kb + 1];
        acc = wmma_f32_k4(a, bb, acc);
      }
      float* dst = affbuf + (c0 + ch) * 1024;
#pragma unroll
      for (int r = 0; r < 8; ++r) {
        const int row = m0 + r + (hi ? 8 : 0);
        const int col = n0 + lane15;
        dst[row * 32 + col] = acc[r] * 0.125f;  // scale = 1/sqrt(64)
      }
    }
    __syncthreads();

    // ---- row softmax over k (32 cols) for both channels ----
    if (tid < 64) {
      const int ch = tid >> 5;
      const int row = tid & 31;
      float* rp = affbuf + (c0 + ch) * 1024 + row * 32;
      float m = rp[0];
      for (int j = 1; j < 32; ++j) m = fmaxf(m, rp[j]);
      float s = 0.0f;
      for (int j = 0; j < 32; ++j) { float e = expf(rp[j] - m); rp[j] = e; s += e; }
      const float inv = 1.0f / s;
      for (int j = 0; j < 32; ++j) rp[j] *= inv;
    }
    __syncthreads();
  }

  // ---- t_conv: 24 -> 1, 3x3 SAME over the 32x32 images ----
  for (int o = tid; o < 1024; o += 256) {
    const int ty = o >> 5, tx = o & 31;
    float s = wsm[216];
    for (int c = 0; c < 24; ++c) {
      const float* ab = affbuf + c * 1024;
      const float* wc = wsm + c * 9;
#pragma unroll
      for (int dy = 0; dy < 3; ++dy) {
        const int y = ty + dy - 1;
        if ((unsigned)y >= 32u) continue;
#pragma unroll
        for (int dx = 0; dx < 3; ++dx) {
          const int x = tx + dx - 1;
          if ((unsigned)x >= 32u) continue;
          s += wc[dy * 3 + dx] * ab[y * 32 + x];
        }
      }
    }
    affout[(size_t)bl * 1024 + o] = s;
  }
}

// ===========================================================================
// Kernel 2: LayerNorm(1024) + MLP(1024->32 GELU ->1024) residual, in place.
// One block per (b,l).
// ===========================================================================
extern "C" __global__ __launch_bounds__(256) void syno_mlp_kernel(
    float* __restrict__ aff, const float* __restrict__ ln_g,
    const float* __restrict__ ln_b, const float* __restrict__ w1,
    const float* __restrict__ b1, const float* __restrict__ w2,
    const float* __restrict__ b2) {
  __shared__ float xs[1024];
  __shared__ float xn[1024];
  __shared__ float red[256];
  __shared__ float part[8][32];
  __shared__ float gact[32];

  const int bl = blockIdx.x;
  const int tid = threadIdx.x;
  float* row = aff + (size_t)bl * 1024;

  float ps = 0.0f, pq = 0.0f;
  for (int i = tid; i < 1024; i += 256) {
    const float vv = row[i];
    xs[i] = vv; ps += vv; pq += vv * vv;
  }
  red[tid] = ps; __syncthreads();
  for (int s = 128; s > 0; s >>= 1) { if (tid < s) red[tid] += red[tid + s]; __syncthreads(); }
  const float mu = red[0] * (1.0f / 1024.0f);
  __syncthreads();
  red[tid] = pq; __syncthreads();
  for (int s = 128; s > 0; s >>= 1) { if (tid < s) red[tid] += red[tid + s]; __syncthreads(); }
  const float var = red[0] * (1.0f / 1024.0f) - mu * mu;
  const float inv = rsqrtf(var + 1e-5f);
  __syncthreads();

  for (int i = tid; i < 1024; i += 256)
    xn[i] = (xs[i] - mu) * inv * ln_g[i] + ln_b[i];
  __syncthreads();

  // h1 = xn @ w1 (1024x32), split K over 8 segments
  const int j = tid & 31, seg = tid >> 5;
  float p = 0.0f;
  for (int i = seg * 128; i < seg * 128 + 128; ++i) p += xn[i] * w1[i * 32 + j];
  part[seg][j] = p;
  __syncthreads();
  if (tid < 32) {
    float h = b1[tid];
#pragma unroll
    for (int s2 = 0; s2 < 8; ++s2) h += part[s2][tid];
    gact[tid] = 0.5f * h * (1.0f + erff(h * 0.70710678118654752f));  // exact GELU
  }
  __syncthreads();

  for (int i = tid; i < 1024; i += 256) {
    float a2 = b2[i];
#pragma unroll
    for (int jj = 0; jj < 32; ++jj) a2 += gact[jj] * w2[jj * 1024 + i];
    row[i] = xs[i] + a2;  // residual, in place
  }
}

// ===========================================================================
// Kernel 3: p_conv 1024->1, 3x3 SAME over 14x14, then flatten. One block
// per output pixel (b,p1,p2).
// ===========================================================================
extern "C" __global__ __launch_bounds__(256) void syno_pconv_kernel(
    const float* __restrict__ aff, const float* __restrict__ pw,
    const float* __restrict__ pb, float* __restrict__ yt) {
  __shared__ float red[256];
  const int idx = blockIdx.x;
  const int b = idx / NL;
  const int pp = idx % NL;
  const int p1 = pp / NP, p2 = pp % NP;
  const int tid = threadIdx.x;

  float acc = 0.0f;
  for (int c = tid; c < 1024; c += 256) {
    const float* wc = pw + c * 9;
#pragma unroll
    for (int dy = 0; dy < 3; ++dy) {
      const int y = p1 + dy - 1;
      if ((unsigned)y >= (unsigned)NP) continue;
#pragma unroll
      for (int dx = 0; dx < 3; ++dx) {
        const int x = p2 + dx - 1;
        if ((unsigned)x >= (unsigned)NP) continue;
        acc += wc[dy * 3 + dx] * aff[((size_t)b * NL + y * NP + x) * 1024 + c];
      }
    }
  }
  red[tid] = acc; __syncthreads();
  for (int s = 128; s > 0; s >>= 1) { if (tid < s) red[tid] += red[tid + s]; __syncthreads(); }
  if (tid == 0) yt[idx] = red[0] + pb[0];
}

// ===========================================================================
// Kernel 4: spatial syno attention. One block per (b,t) (512 blocks).
// GEMM1: syno(16x768) @ kk^T -> (16x196), softmax, GEMM2: @ vv -> (16x768),
// sum over 16 synos, atomic accumulate mean into y_s.
// ===========================================================================
#define KCH 32
#define KPAD 36
#define APAD 210

extern "C" __global__ __launch_bounds__(256) void syno_spatial_kernel(
    const float* __restrict__ kin, const float* __restrict__ vin,
    const float* __restrict__ syno, float* __restrict__ ys) {
  __shared__ float kch[NL * KPAD];   // 196*36
  __shared__ float sch[NS * KPAD];   // 16*36
  __shared__ float Abuf[NS * APAD];  // 16*210

  const int bt = blockIdx.x;
  const int b = bt >> 5;
  const int t = bt & 31;
  const int tid = threadIdx.x;
  const int lane = tid & 31;
  const int wave = tid >> 5;
  const int lane15 = lane & 15;
  const int hi = lane >> 4;
  const int koffBase = hi ? 2 : 0;

  const float* kbase = kin + ((size_t)(b * NT + t) * (NL + 1) + 1) * NDM;
  const float* vbase = vin + ((size_t)(b * NT + t) * (NL + 1) + 1) * NDM;
  const float synoscale = 0.03608439182435161f;  // 1/sqrt(768)

  // -------- GEMM1: A = syno_scaled @ kk^T, 13 N-tiles over 8 waves --------
  v8f accA0 = {};
  v8f accA1 = {};
  const int n0a = wave * 16;
  const int n0b = (wave + 8) * 16;  // only valid for wave < 5 (tiles 8..12)

  for (int kc = 0; kc < NDM; kc += KCH) {
    for (int i = tid; i < NS * KCH; i += 256) {
      const int s = i >> 5, kk = i & 31;
      sch[s * KPAD + kk] = syno[s * NDM + kc + kk] * synoscale;
    }
    for (int i = tid; i < NL * KCH; i += 256) {
      const int l = i >> 5, kk = i & 31;
      kch[l * KPAD + kk] = kbase[(size_t)l * NDM + kc + kk];
    }
    __syncthreads();

    const float* arow = sch + lane15 * KPAD + koffBase;
    {
      int n = n0a + lane15; if (n > NL - 1) n = NL - 1;
      const float* brow = kch + n * KPAD + koffBase;
#pragma unroll
      for (int kb = 0; kb < KCH; kb += 4) {
        v2f a;  a.x = arow[kb]; a.y = arow[kb + 1];
        v2f bb; bb.x = brow[kb]; bb.y = brow[kb + 1];
        accA0 = wmma_f32_k4(a, bb, accA0);
      }
    }
    if (wave < 5) {  // wave-uniform branch: EXEC stays all-ones inside
      int n = n0b + lane15; if (n > NL - 1) n = NL - 1;
      const float* brow = kch + n * KPAD + koffBase;
#pragma unroll
      for (int kb = 0; kb < KCH; kb += 4) {
        v2f a;  a.x = arow[kb]; a.y = arow[kb + 1];
        v2f bb; bb.x = brow[kb]; bb.y = brow[kb + 1];
        accA1 = wmma_f32_k4(a, bb, accA1);
      }
    }
    __syncthreads();
  }

  // write A tiles to LDS (cols >= 196 get -inf so softmax ignores them)
#pragma unroll
  for (int r = 0; r < 8; ++r) {
    const int row = r + (hi ? 8 : 0);
    Abuf[row * APAD + n0a + lane15] = accA0[r];
  }
  if (wave < 5) {
#pragma unroll
    for (int r = 0; r < 8; ++r) {
      const int row = r + (hi ? 8 : 0);
      const int col = n0b + lane15;
      Abuf[row * APAD + col] = (col < NL) ? accA1[r] : -INFINITY;
    }
  }
  __syncthreads();

  // -------- softmax over the 196 keys, one syno row per thread --------
  if (tid < NS) {
    float* rp = Abuf + tid * APAD;
    float m = -INFINITY;
    for (int jv = 0; jv < NL; ++jv) m = fmaxf(m, rp[jv]);
    float s = 0.0f;
    for (int jv = 0; jv < NL; ++jv) { float e = expf(rp[jv] - m); rp[jv] = e; s += e; }
    const float inv = 1.0f / s;
    for (int jv = 0; jv < NL; ++jv) rp[jv] *= inv;
  }
  __syncthreads();

  // -------- GEMM2: s_mix = P(16x196) @ vv(196x768); 48 N-tiles, 6/wave ----
  const float* arow = Abuf + lane15 * APAD;
  for (int tn = wave; tn < 48; tn += 8) {
    const int n0 = tn * 16;
    v8f acc = {};
    const float* vcol = vbase + n0 + lane15;
#pragma unroll 7
    for (int ks = 0; ks < NL; ks += 4) {
      const int r0 = ks + koffBase;
      v2f a;  a.x = arow[r0]; a.y = arow[r0 + 1];
      v2f bb; bb.x = vcol[(size_t)r0 * NDM]; bb.y = vcol[(size_t)(r0 + 1) * NDM];
      acc = wmma_f32_k4(a, bb, acc);
    }
    // sum over the 16 syno rows: lanes 0-15 hold rows 0-7, 16-31 rows 8-15
    float ptl = acc[0] + acc[1] + acc[2] + acc[3] + acc[4] + acc[5] + acc[6] + acc[7];
    const float tot = ptl + __shfl_xor(ptl, 16, 32);
    if (hi == 0)
      atomicAdd(&ys[b * NDM + n0 + lane15], tot * (1.0f / 512.0f));  // mean over T*S
  }
}

// ===========================================================================
// Host launcher
// ===========================================================================
extern "C" void kernel_launch(void* const* d_in, const int* in_sizes, int n_in,
                              void* d_out, int out_size, void* d_ws,
                              size_t ws_size, hipStream_t stream) {
  (void)in_sizes; (void)n_in; (void)out_size; (void)ws_size;
  const float* q        = (const float*)d_in[0];
  const float* k        = (const float*)d_in[1];
  const float* v        = (const float*)d_in[2];
  const float* t_conv_w = (const float*)d_in[3];
  const float* t_conv_b = (const float*)d_in[4];
  const float* ln_g     = (const float*)d_in[5];
  const float* ln_b     = (const float*)d_in[6];
  const float* w1       = (const float*)d_in[7];
  const float* b1       = (const float*)d_in[8];
  const float* w2       = (const float*)d_in[9];
  const float* b2       = (const float*)d_in[10];
  const float* p_conv_w = (const float*)d_in[11];
  const float* p_conv_b = (const float*)d_in[12];
  const float* syno     = (const float*)d_in[13];

  float* out = (float*)d_out;
  float* yt = out;              // (16,196)
  float* ys = out + NB * NL;    // (16,768)
  float* aff = (float*)d_ws;    // (B*L, 1024) fp32 scratch, ~12.3 MB

  // y_s is accumulated with atomics -> zero it (memset is graph-capturable)
  hipMemsetAsync(ys, 0, (size_t)NB * NDM * sizeof(float), stream);

  const size_t smem1 = (size_t)(24 * 32 * 32 + 2 * 32 * XPAD + 224) * sizeof(float);
  syno_temporal_kernel<<<NB * NL, 256, smem1, stream>>>(q, k, t_conv_w, t_conv_b, aff);
  syno_mlp_kernel<<<NB * NL, 256, 0, stream>>>(aff, ln_g, ln_b, w1, b1, w2, b2);
  syno_pconv_kernel<<<NB * NL, 256, 0, stream>>>(aff, p_conv_w, p_conv_b, yt);
  syno_spatial_kernel<<<NB * NT, 256, 0, stream>>>(k, v, syno, ys);
}